// KNNWRMFSampler_19576460935750
// MI455X (gfx1250) — compile-verified
//
#include <hip/hip_runtime.h>

#ifndef __has_builtin
#define __has_builtin(x) 0
#endif

#define NN      100   // NUM_NEAREST
#define KU      32    // K uniforms per row
#define WPB     8     // wave32s (rows) per block
#define NPAD    128   // cum row padded to power of two for branchless search
#define STRIDE  132   // padded LDS row stride (dwords)
#define BIGF    3.0e38f

#if __has_builtin(__builtin_amdgcn_global_load_async_to_lds_b32) && \
    __has_builtin(__builtin_amdgcn_s_wait_asynccnt)
#define USE_ASYNC 1
#else
#define USE_ASYNC 0
#endif

typedef __attribute__((address_space(1))) int gbl_i32_t;   // "__device__ int*"
typedef __attribute__((address_space(3))) int lds_i32_t;   // "__shared__ int*"

__device__ __forceinline__ void async_b32(const void* g, void* l) {
#if USE_ASYNC
  __builtin_amdgcn_global_load_async_to_lds_b32((gbl_i32_t*)g, (lds_i32_t*)l,
                                                /*imm offset*/0, /*cpol*/0);
#else
  *(int*)l = *(const int*)g;
#endif
}

__global__ __launch_bounds__(WPB * 32)
void knn_wrmf_sampler_kernel(const int*   __restrict__ trg_seq,     // (L,2)
                             const float* __restrict__ uniforms,    // (L,K)
                             const int*   __restrict__ knn_results, // (N_LOCS,NN)
                             const float* __restrict__ probs_table, // (N_LOCS+1,NN)
                             const float* __restrict__ cum_table,   // (N_LOCS+1,NN)
                             float*       __restrict__ out,         // [L*K ns][L*K sp]
                             int L)
{
  __shared__ float cum_s [WPB][STRIDE];
  __shared__ float prob_s[WPB][STRIDE];
  __shared__ int   near_s[WPB][STRIDE];

  const int lane = threadIdx.x & 31;
  const int wave = threadIdx.x >> 5;
  const int row  = blockIdx.x * WPB + wave;
  if (row >= L) return;

  // trg_loc in [1, N_LOCS] is wave-uniform: force scalar so the row bases
  // live in SGPRs and async loads use saddr + per-lane voffset.
  const int trg_loc = __builtin_amdgcn_readfirstlane(trg_seq[2 * row + 1]);
  const float* cum_row  = cum_table   + (long)trg_loc * NN;
  const float* prob_row = probs_table + (long)trg_loc * NN;
  const int*   knn_row  = knn_results + (long)(trg_loc - 1) * NN;

  // ---- stage the three 100-element rows into this wave's LDS slice ----
  // j0..j2 cover 0..95; j3 = 96+(lane&3): 8 lanes redundantly fetch each of
  // 96..99 (same data -> benign duplicate LDS writes, no divergence).
  const int j0 = lane, j1 = lane + 32, j2 = lane + 64, j3 = 96 + (lane & 3);

  async_b32(cum_row  + j0, &cum_s [wave][j0]);
  async_b32(cum_row  + j1, &cum_s [wave][j1]);
  async_b32(cum_row  + j2, &cum_s [wave][j2]);
  async_b32(cum_row  + j3, &cum_s [wave][j3]);

  async_b32(prob_row + j0, &prob_s[wave][j0]);
  async_b32(prob_row + j1, &prob_s[wave][j1]);
  async_b32(prob_row + j2, &prob_s[wave][j2]);
  async_b32(prob_row + j3, &prob_s[wave][j3]);

  async_b32(knn_row  + j0, &near_s[wave][j0]);
  async_b32(knn_row  + j1, &near_s[wave][j1]);
  async_b32(knn_row  + j2, &near_s[wave][j2]);
  async_b32(knn_row  + j3, &near_s[wave][j3]);

  // pad cum row to 128 with +BIG so the 7-step search needs no bound checks
  if (lane < NPAD - NN) cum_s[wave][NN + lane] = BIGF;

#if USE_ASYNC
  __builtin_amdgcn_s_wait_asynccnt(0);
#endif
  __builtin_amdgcn_wave_barrier();   // LDS fill before LDS reads

  // ---- branchless bitwise lower_bound over 128 padded entries ----
  // pos = #elements < u  (searchsorted side='left')
  const float u = uniforms[(long)row * KU + lane];
  const float* c = cum_s[wave];
  int pos = 0;
#pragma unroll
  for (int s = 64; s >= 1; s >>= 1)
    pos += (c[pos + s - 1] < u) ? s : 0;
  const int idx = (pos > NN - 1) ? (NN - 1) : pos;

  // ---- gather sample + prob from LDS, coalesced store ----
  const long o = (long)row * KU + lane;
  out[o]                = (float)near_s[wave][idx]; // neg_samples (exact in f32)
  out[(long)L * KU + o] = prob_s[wave][idx];        // sample_probs
}

extern "C" void kernel_launch(void* const* d_in, const int* in_sizes, int n_in,
                              void* d_out, int out_size, void* d_ws, size_t ws_size,
                              hipStream_t stream) {
  // setup_inputs order: trg_seq, k, user, uniforms, knn_results, probs_table, cum_probs_table
  const int*   trg_seq  = (const int*)  d_in[0];
  const float* uniforms = (const float*)d_in[3];
  const int*   knn      = (const int*)  d_in[4];
  const float* probs    = (const float*)d_in[5];
  const float* cum      = (const float*)d_in[6];
  float* out = (float*)d_out;

  const int L = in_sizes[0] / 2;                 // 4096
  const int blocks = (L + WPB - 1) / WPB;        // 512 blocks x 256 threads
  hipLaunchKernelGGL(knn_wrmf_sampler_kernel, dim3(blocks), dim3(WPB * 32), 0, stream,
                     trg_seq, uniforms, knn, probs, cum, out, L);
}